// TrajCostMPC_8400956031312
// MI455X (gfx1250) — compile-verified
//
#include <hip/hip_runtime.h>

// iLQR (T=30, B=32768, N=4, M=2, NM=6) for MI455X / gfx1250 — fused kernel,
// 3-deep async pipeline. TERMINAL DESIGN.
//
// Roofline: ~1.2 GFLOP vs ~575 MB HBM traffic -> hard bandwidth-bound
// (~25 us floor @ 23.3 TB/s vs ~1 us fp32 VALU). The CDNA5 feature that pays
// is the async data-movement path, not WMMA (6x6 fp32 tiles fill <=72/256 of
// v_wmma_f32_16x16x4_f32; chaining needs cross-lane repacks). Design:
//  - ONE fused kernel: forward(block i) depends only on backward(block i);
//    K,k gains (30 x 32 x 12 floats = 45 KB) live entirely in LDS.
//  - 32-thread blocks = single wave32 (1-wave WG barriers are S_NOP; no
//    __syncthreads needed), one batch element per thread.
//  - 3 staging buffers, prefetch depth 2: ~18 KB per wave in flight, two
//    compute phases of HBM latency tolerance. Partial s_wait_asynccnt
//    immediates exploit per-wave in-order async completion (full slab = 19
//    ops: 16 b128 + 3 b64; C/c-only slab = 12 ops). s_wait_dscnt 0 guards
//    buffer re-targeting (DS vs async-LDS may complete out of order).
//  - dual-sided 24-bit INST_OFFSET immediates: one address pair per slab.
//  - L2 policy: backward RT (retains t~0..20 slabs read last, which forward
//    reads first), forward NT (last use), NT stores for outputs.

#define TT   30
#define BB   32768
#define NN   4
#define MM   2
#define NMD  6
#define BT   32          // one wave32 per block, one batch element per thread
#define REGU 1e-6f

typedef float v4f __attribute__((ext_vector_type(4)));
typedef float v2f __attribute__((ext_vector_type(2)));

// ---------------- CDNA5 async global->LDS staging ----------------
__device__ __forceinline__ unsigned lds_u32(const void* p) {
  return (unsigned)(unsigned long long)p;
}
template<int N>
__device__ __forceinline__ void wait_async_le() {
  asm volatile("s_wait_asynccnt %0" :: "n"(N) : "memory");
}
__device__ __forceinline__ void wait_ds0() {
  asm volatile("s_wait_dscnt 0" ::: "memory");
}

// Single async copy; compile-time immediate offset applies to BOTH the global
// source and the LDS destination. TH: 0 = RT, 1 = NT.
template<int TH, int OFF>
__device__ __forceinline__ void a16(unsigned lo, const float* g) {
  if constexpr (TH == 0)
    asm volatile("global_load_async_to_lds_b128 %0, %1, off offset:%2"
                 :: "v"(lo), "v"(g), "n"(OFF) : "memory");
  else
    asm volatile("global_load_async_to_lds_b128 %0, %1, off offset:%2 th:TH_LOAD_NT"
                 :: "v"(lo), "v"(g), "n"(OFF) : "memory");
}
template<int TH, int OFF>
__device__ __forceinline__ void a8(unsigned lo, const float* g) {
  if constexpr (TH == 0)
    asm volatile("global_load_async_to_lds_b64 %0, %1, off offset:%2"
                 :: "v"(lo), "v"(g), "n"(OFF) : "memory");
  else
    asm volatile("global_load_async_to_lds_b64 %0, %1, off offset:%2 th:TH_LOAD_NT"
                 :: "v"(lo), "v"(g), "n"(OFF) : "memory");
}

// C++17 compile-time unrollers so OFF is a genuine immediate.
template<int J, int N, int TH>
struct Rep16 {
  static __device__ __forceinline__ void go(unsigned lo, const float* g) {
    a16<TH, J * BT * 16>(lo, g);
    Rep16<J + 1, N, TH>::go(lo, g);
  }
};
template<int N, int TH>
struct Rep16<N, N, TH> {
  static __device__ __forceinline__ void go(unsigned, const float*) {}
};
template<int J, int N, int TH>
struct Rep8 {
  static __device__ __forceinline__ void go(unsigned lo, const float* g) {
    a8<TH, J * BT * 8>(lo, g);
    Rep8<J + 1, N, TH>::go(lo, g);
  }
};
template<int N, int TH>
struct Rep8<N, N, TH> {
  static __device__ __forceinline__ void go(unsigned, const float*) {}
};

template<int NCH, int TH>
__device__ __forceinline__ void stage16(float* dst, const float* src, int tid) {
  Rep16<0, NCH, TH>::go(lds_u32(dst) + (unsigned)tid * 16u,
                        src + (size_t)tid * 4);
}
template<int NCH, int TH>
__device__ __forceinline__ void stage8(float* dst, const float* src, int tid) {
  Rep8<0, NCH, TH>::go(lds_u32(dst) + (unsigned)tid * 8u,
                       src + (size_t)tid * 2);
}

// Staging buffer layout (floats): C[36*BT] | F[24*BT] | c[6*BT] | f[4*BT]
#define SB_C 0
#define SB_F (BT * 36)
#define SB_c (BT * 60)
#define SB_f (BT * 66)
#define SB_TOT (BT * 70)
#define NBUF 3
#define G_OFF (NBUF * SB_TOT)              // gains region (floats)
#define G_STRIDE 12                        // 8 (K) + 2 (k) + 2 pad -> 48B, 16B-aligned
#define LDS_FLOATS (G_OFF + TT * BT * G_STRIDE)   // 18,240 floats = 72,960 B

// Slab pieces: C+c = 12 async ops, +F+f = 19 async ops.
template<int TH>
__device__ __forceinline__ void stage_Cc(float* buf, const float* Cg,
                                         const float* cg, size_t o, int tid) {
  stage16<9, TH>(buf + SB_C, Cg + o * 36, tid);
  stage8 <3, TH>(buf + SB_c, cg + o * 6,  tid);
}
template<int TH>
__device__ __forceinline__ void stage_Ff(float* buf, const float* Fg,
                                         const float* fg, size_t o, int tid) {
  stage16<6, TH>(buf + SB_F, Fg + o * 24, tid);
  stage16<1, TH>(buf + SB_f, fg + o * 4,  tid);
}

__global__ void __launch_bounds__(BT) ilqr_fused(
    const float* __restrict__ xg, const float* __restrict__ Cg,
    const float* __restrict__ cg, const float* __restrict__ Fg,
    const float* __restrict__ fg, float* __restrict__ xs,
    float* __restrict__ us, float* __restrict__ cost_out)
{
  extern __shared__ float smem[];
  float* gains = smem + G_OFF;
  const int tid   = threadIdx.x;
  const int bbase = blockIdx.x * BT;
  const int b     = bbase + tid;

  // Issue the x_init load now: its latency hides under phase-1 staging.
  float x[NN];
  {
    const v4f xi = *(const v4f*)(xg + (size_t)b * 4);
    x[0] = xi.x; x[1] = xi.y; x[2] = xi.z; x[3] = xi.w;
  }

  // ================= phase 1: backward Riccati =================
  float V[NN][NN], v[NN];
#pragma unroll
  for (int i = 0; i < NN; ++i) {
    v[i] = 0.f;
#pragma unroll
    for (int j = 0; j < NN; ++j) V[i][j] = 0.f;
  }

  // Prologue: stage slabs for t = 29 (C,c only: F,f are the zero pad),
  // t = 28, t = 27 into buffers 0,1,2.
  stage_Cc<0>(smem,               Cg, cg, (size_t)(TT - 1) * BB + bbase, tid);
  stage_Cc<0>(smem + SB_TOT,      Cg, cg, (size_t)(TT - 2) * BB + bbase, tid);
  stage_Ff<0>(smem + SB_TOT,      Fg, fg, (size_t)(TT - 2) * BB + bbase, tid);
  stage_Cc<0>(smem + 2 * SB_TOT,  Cg, cg, (size_t)(TT - 3) * BB + bbase, tid);
  stage_Ff<0>(smem + 2 * SB_TOT,  Fg, fg, (size_t)(TT - 3) * BB + bbase, tid);

  for (int t = TT - 1; t >= 0; --t) {
    // Wait until this step's slab landed: async completes in order, so wait
    // down to the op count of the (up to 2) newer slabs still in flight.
    if (t >= 2)      wait_async_le<38>();
    else if (t == 1) wait_async_le<19>();
    else             wait_async_le<0>();

    float* buf = smem + ((TT - 1 - t) % NBUF) * SB_TOT;
    const float* Cl = buf + SB_C + tid * 36;
    const float* cl = buf + SB_c + tid * 6;
    const float* Fl = buf + SB_F + tid * 24;
    const float* fl = buf + SB_f + tid * 4;

    float Q[NMD * NMD], q[NMD];
    if (t == TT - 1) {                 // padded last step: F = 0, f = 0
#pragma unroll
      for (int i = 0; i < NMD * NMD; ++i) Q[i] = Cl[i];
#pragma unroll
      for (int i = 0; i < NMD; ++i) q[i] = cl[i];
    } else {
      float Fr[NN][NMD];
#pragma unroll
      for (int n = 0; n < NN; ++n)
#pragma unroll
        for (int j = 0; j < NMD; ++j) Fr[n][j] = Fl[n * NMD + j];

      float VF[NN][NMD];                       // VF = V @ F
#pragma unroll
      for (int n = 0; n < NN; ++n)
#pragma unroll
        for (int j = 0; j < NMD; ++j) {
          float s = 0.f;
#pragma unroll
          for (int m = 0; m < NN; ++m) s += V[n][m] * Fr[m][j];
          VF[n][j] = s;
        }
#pragma unroll
      for (int i = 0; i < NMD; ++i)            // Q = C + F^T VF
#pragma unroll
        for (int j = 0; j < NMD; ++j) {
          float s = Cl[i * NMD + j];
#pragma unroll
          for (int n = 0; n < NN; ++n) s += Fr[n][i] * VF[n][j];
          Q[i * NMD + j] = s;
        }
      float w[NN];                             // w = V f + v
#pragma unroll
      for (int n = 0; n < NN; ++n) {
        float s = v[n];
#pragma unroll
        for (int m = 0; m < NN; ++m) s += V[n][m] * fl[m];
        w[n] = s;
      }
#pragma unroll
      for (int i = 0; i < NMD; ++i) {          // q = c + F^T w
        float s = cl[i];
#pragma unroll
        for (int n = 0; n < NN; ++n) s += Fr[n][i] * w[n];
        q[i] = s;
      }
    }

    // K = -(Quu + REG I)^-1 [Qux | qu]   (2x2 explicit inverse)
    const float a00 = Q[4 * NMD + 4] + REGU, a01 = Q[4 * NMD + 5];
    const float a10 = Q[5 * NMD + 4],        a11 = Q[5 * NMD + 5] + REGU;
    const float inv = 1.0f / (a00 * a11 - a01 * a10);
    float K[MM][NN], kk[MM];
#pragma unroll
    for (int j = 0; j < NN; ++j) {
      const float r0 = Q[4 * NMD + j], r1 = Q[5 * NMD + j];
      K[0][j] = -(a11 * r0 - a01 * r1) * inv;
      K[1][j] = -(a00 * r1 - a10 * r0) * inv;
    }
    {
      const float r0 = q[4], r1 = q[5];
      kk[0] = -(a11 * r0 - a01 * r1) * inv;
      kk[1] = -(a00 * r1 - a10 * r0) * inv;
    }

    // Vn = Qxx + Qxu K + K^T Qux + K^T Quu K ; vn = qx + Qxu k + K^T(qu + Quu k)
    float Vn[NN][NN], vn[NN];
#pragma unroll
    for (int i = 0; i < NN; ++i)
#pragma unroll
      for (int j = 0; j < NN; ++j) {
        float s = Q[i * NMD + j];
#pragma unroll
        for (int u = 0; u < MM; ++u) s += Q[i * NMD + 4 + u] * K[u][j];
#pragma unroll
        for (int u = 0; u < MM; ++u) s += K[u][i] * Q[(4 + u) * NMD + j];
#pragma unroll
        for (int u = 0; u < MM; ++u) {
          float acc = 0.f;
#pragma unroll
          for (int w2 = 0; w2 < MM; ++w2) acc += Q[(4 + u) * NMD + 4 + w2] * K[w2][j];
          s += K[u][i] * acc;
        }
        Vn[i][j] = s;
      }
#pragma unroll
    for (int i = 0; i < NN; ++i) {
      float s = q[i];
#pragma unroll
      for (int u = 0; u < MM; ++u) s += Q[i * NMD + 4 + u] * kk[u];
#pragma unroll
      for (int u = 0; u < MM; ++u) {
        float t2 = q[4 + u];
#pragma unroll
        for (int w2 = 0; w2 < MM; ++w2) t2 += Q[(4 + u) * NMD + 4 + w2] * kk[w2];
        s += K[u][i] * t2;
      }
      vn[i] = s;
    }
#pragma unroll
    for (int i = 0; i < NN; ++i) {
      v[i] = vn[i];
#pragma unroll
      for (int j = 0; j < NN; ++j) V[i][j] = 0.5f * (Vn[i][j] + Vn[j][i]);
    }

    // Gains stay in LDS (48B stride: 16B-aligned, conflict-free for wave32).
    float* gl = gains + (size_t)(t * BT + tid) * G_STRIDE;
    *(v4f*)(gl + 0) = (v4f){K[0][0], K[0][1], K[0][2], K[0][3]};
    *(v4f*)(gl + 4) = (v4f){K[1][0], K[1][1], K[1][2], K[1][3]};
    *(v2f*)(gl + 8) = (v2f){kk[0], kk[1]};

    // Refill this buffer with the slab for step t-3 (depth-2 prefetch).
    if (t >= 3) {
      wait_ds0();                      // this buffer's reads are complete
      const size_t o = (size_t)(t - 3) * BB + (size_t)bbase;
      stage_Cc<0>(buf, Cg, cg, o, tid);
      stage_Ff<0>(buf, Fg, fg, o, tid);
    }
  }

  // ================= phase 2: forward rollout =================
  float cost = 0.f;

  wait_ds0();                          // phase-1 reads of smem are complete
  // Prologue: slabs t = 0,1,2 into buffers 0,1,2 (NT: last use of inputs).
  stage_Cc<1>(smem,              Cg, cg, (size_t)0 * BB + bbase, tid);
  stage_Ff<1>(smem,              Fg, fg, (size_t)0 * BB + bbase, tid);
  stage_Cc<1>(smem + SB_TOT,     Cg, cg, (size_t)1 * BB + bbase, tid);
  stage_Ff<1>(smem + SB_TOT,     Fg, fg, (size_t)1 * BB + bbase, tid);
  stage_Cc<1>(smem + 2 * SB_TOT, Cg, cg, (size_t)2 * BB + bbase, tid);
  stage_Ff<1>(smem + 2 * SB_TOT, Fg, fg, (size_t)2 * BB + bbase, tid);

  for (int t = 0; t < TT; ++t) {
    // Newer slabs in flight: t+1,t+2 (full=19 each; slab 29 is C,c only=12).
    if (t <= 26)      wait_async_le<38>();
    else if (t == 27) wait_async_le<31>();
    else if (t == 28) wait_async_le<12>();
    else              wait_async_le<0>();

    float* buf = smem + (t % NBUF) * SB_TOT;
    const float* Cl = buf + SB_C + tid * 36;
    const float* cl = buf + SB_c + tid * 6;
    const float* Fl = buf + SB_F + tid * 24;
    const float* fl = buf + SB_f + tid * 4;
    const float* gl = gains + (size_t)(t * BT + tid) * G_STRIDE;

    float tau[NMD];
#pragma unroll
    for (int i = 0; i < NN; ++i) tau[i] = x[i];
#pragma unroll
    for (int m = 0; m < MM; ++m) {     // u = K x + k
      float s = gl[8 + m];
#pragma unroll
      for (int n = 0; n < NN; ++n) s += gl[m * NN + n] * x[n];
      tau[NN + m] = s;
    }

    float cs = 0.f;                    // 0.5 tau^T C tau + tau . c
#pragma unroll
    for (int i = 0; i < NMD; ++i) {
      float s = 0.f;
#pragma unroll
      for (int j = 0; j < NMD; ++j) s += Cl[i * NMD + j] * tau[j];
      cs += tau[i] * (0.5f * s + cl[i]);
    }
    cost += cs;

    // Streaming outputs: never re-read on device -> non-temporal stores.
    const size_t o = (size_t)t * BB + (size_t)b;
    __builtin_nontemporal_store((v4f){x[0], x[1], x[2], x[3]},
                                (v4f*)(xs + o * 4));
    __builtin_nontemporal_store((v2f){tau[4], tau[5]},
                                (v2f*)(us + o * 2));

    if (t < TT - 1) {                  // x' = F tau + f
      float xn[NN];
#pragma unroll
      for (int n = 0; n < NN; ++n) {
        float s = fl[n];
#pragma unroll
        for (int i = 0; i < NMD; ++i) s += Fl[n * NMD + i] * tau[i];
        xn[n] = s;
      }
#pragma unroll
      for (int n = 0; n < NN; ++n) x[n] = xn[n];
    }

    // Refill this buffer with slab t+3 (F,f exist only for slab < T-1).
    if (t + 3 <= TT - 1) {
      wait_ds0();
      const size_t o2 = (size_t)(t + 3) * BB + (size_t)bbase;
      stage_Cc<1>(buf, Cg, cg, o2, tid);
      if (t + 3 < TT - 1) stage_Ff<1>(buf, Fg, fg, o2, tid);
    }
  }
  __builtin_nontemporal_store(cost, cost_out + b);
}

// ---------------- host launch ----------------
extern "C" void kernel_launch(void* const* d_in, const int* in_sizes, int n_in,
                              void* d_out, int out_size, void* d_ws, size_t ws_size,
                              hipStream_t stream) {
  const float* x_init = (const float*)d_in[0];
  const float* C      = (const float*)d_in[1];
  const float* c      = (const float*)d_in[2];
  const float* F      = (const float*)d_in[3];
  const float* f      = (const float*)d_in[4];

  float* out = (float*)d_out;
  float* xs  = out;
  float* us  = out + (size_t)TT * BB * 4;
  float* co  = out + (size_t)TT * BB * 6;

  dim3 grid(BB / BT), block(BT);
  ilqr_fused<<<grid, block, (size_t)LDS_FLOATS * sizeof(float), stream>>>(
      x_init, C, c, F, f, xs, us, co);
}